// fusion_class_layer_42691974922700
// MI455X (gfx1250) — compile-verified
//
#include <hip/hip_runtime.h>
#include <cstdint>

// Problem shape (fixed by the reference).
#define BATCH 32768
#define NCLS  400
#define NMOD  4

// Tile: 128 batches x 16 classes per 256-thread block.
// 32768/128 = 256, 400/16 = 25 -> no edge cases.
#define BT 128
#define CT 16
#define ROWP 129  // row pitch in float4 units (128 + 1 pad -> conflict-free ds_load_b128)

// The async-to-LDS builtin takes int4 pointers (per hipcc's diagnostic):
//   param0: int4 addrspace(1)* (global src), param1: int4 addrspace(3)* (LDS dst)
typedef int v4i __attribute__((vector_size(16)));
typedef __attribute__((address_space(1))) v4i gv4i;
typedef __attribute__((address_space(3))) v4i lv4i;
typedef __attribute__((address_space(3))) char lds_char;

__global__ __launch_bounds__(256) void
fusion_class_layer_kernel(const float4* __restrict__ L,   // [B, C] of float4 (M=4)
                          const float4* __restrict__ W,   // [C]    of float4
                          float* __restrict__ out)        // [C, B]
{
    __shared__ float4 tile[CT * ROWP];   // tile[c][b], padded rows (~33 KB)

    const int t  = threadIdx.x;          // 0..255
    const int b0 = blockIdx.x * BT;      // batch tile origin
    const int c0 = blockIdx.y * CT;      // class tile origin

    // ---- Stage 1: async global -> LDS, transpose folded into the LDS address.
    // Global side: for fixed b, 16 consecutive lanes read 16 consecutive float4
    // (256 B contiguous, fully-used cachelines). LDS side: scatter to tile[c][b].
    #pragma unroll
    for (int i = 0; i < (BT * CT) / 256; ++i) {           // 8 async b128 per thread
        const int li = i * 256 + t;
        const int c  = li & (CT - 1);                     // li % 16
        const int b  = li >> 4;                           // li / 16, 0..127
        const float4* gp = L + ((size_t)(b0 + b) * NCLS + (size_t)(c0 + c));
        float4*       lp = &tile[c * ROWP + b];
#if __has_builtin(__builtin_amdgcn_global_load_async_to_lds_b128)
        __builtin_amdgcn_global_load_async_to_lds_b128(
            (gv4i*)gp, (lv4i*)lp, /*offset=*/0, /*cpol=*/0);
#else
        const uint32_t loff = (uint32_t)(uintptr_t)(lds_char*)(char*)lp;
        const uint64_t ga   = (uint64_t)(uintptr_t)gp;
        asm volatile("global_load_async_to_lds_b128 %0, %1, off"
                     :: "v"(loff), "v"(ga) : "memory");
#endif
    }

    // Drain this wave's ASYNCcnt, then make the tile visible workgroup-wide.
#if __has_builtin(__builtin_amdgcn_s_wait_asynccnt)
    __builtin_amdgcn_s_wait_asynccnt(0);
#else
    asm volatile("s_wait_asynccnt 0x0" ::: "memory");
#endif
    __syncthreads();

    // ---- Stage 2: dot(L[b,c,:], W[c]) from LDS, coalesced transposed stores.
    // Thread t -> b = t % 128 (contiguous across a wave), c_grp = t / 128.
    // c is wave-uniform -> W loads scalarize; stores are 512 B contiguous per class row.
    const int b     = t & (BT - 1);
    const int c_grp = t >> 7;                             // 0 or 1
    #pragma unroll
    for (int j = 0; j < CT / 2; ++j) {
        const int    c = c_grp * (CT / 2) + j;
        const float4 v = tile[c * ROWP + b];              // ds_load_b128, conflict-free
        const float4 w = W[c0 + c];                       // tiny, L0/L2-resident, uniform
        const float acc = v.x * w.x + v.y * w.y + v.z * w.z + v.w * w.w;
        out[(size_t)(c0 + c) * BATCH + (size_t)(b0 + b)] = acc;
    }
}

extern "C" void kernel_launch(void* const* d_in, const int* in_sizes, int n_in,
                              void* d_out, int out_size, void* d_ws, size_t ws_size,
                              hipStream_t stream) {
    (void)in_sizes; (void)n_in; (void)out_size; (void)d_ws; (void)ws_size;
    const float4* L  = (const float4*)d_in[0];   // [32768, 400, 4] f32 == [32768,400] float4
    const float4* W  = (const float4*)d_in[1];   // [400, 4] f32 == [400] float4
    float*       out = (float*)d_out;            // [400, 32768, 1] f32

    dim3 grid(BATCH / BT, NCLS / CT);            // (256, 25)
    fusion_class_layer_kernel<<<grid, 256, 0, stream>>>(L, W, out);
}